// SAG_4861902979729
// MI455X (gfx1250) — compile-verified
//
#include <hip/hip_runtime.h>

// CDNA5 (gfx1250) wave32 WMMA types
typedef __attribute__((ext_vector_type(2))) float v2f;   // A/B for 16x16x4 f32: 2 VGPRs
typedef __attribute__((ext_vector_type(8))) float v8f;   // C/D 16x16 f32: 8 VGPRs
typedef __attribute__((ext_vector_type(4))) int   v4i;   // one b128 load = 2 int64 indices

// ---------------------------------------------------------------------------
// Fast path: uniform degree 32, N % 16 == 0, all offsets fit in 32 bits.
// X_out[i] = sum_{k<32} X[col[i*32+k]] as D(16x16) += A_sel(16x4) @ B(4x16)
// via V_WMMA_F32_16X16X4_F32 (exact f32). One wave = 16 rows x 16 features.
// All memory ops use uniform SGPR base + unsigned 32-bit VGPR offset so the
// compiler can select GVS (saddr) addressing instead of 64-bit VALU chains.
// ---------------------------------------------------------------------------
__global__ __launch_bounds__(256) void sag_wmma_deg32_kernel(
    const float* __restrict__ X,
    const int*   __restrict__ col2,   // int64 column_index viewed as int pairs
    float* __restrict__ out,
    int D, int FB, int tiles)
{
    const int lane = threadIdx.x & 31;
    const int wave = threadIdx.x >> 5;
    const int tile = blockIdx.x * 8 + wave;
    if (tile >= tiles) return;              // whole-wave uniform exit

    const int rb   = tile / FB;
    const int fb   = tile % FB;
    const int row0 = rb * 16;
    const int n0   = fb * 16;

    const int lh = lane >> 4;               // lane half: selects K pair / M half
    const int n  = lane & 15;               // feature column within tile
    const int nf = n0 + n;
    const int nok = (nf < D);
    // Clamp the feature offset only: garbage in B columns with nf>=D stays in
    // D columns we never store (WMMA column n reads only B column n).
    const unsigned ncl = (unsigned)(nok ? nf : 0);

    v8f acc0 = {};                          // even K-chunks
    v8f acc1 = {};                          // odd  K-chunks (breaks dep chain)

    for (int m = 0; m < 16; ++m) {
        // Selector A: ones only in output row m (exact in f32)
        const float sel = (n == m) ? 1.0f : 0.0f;
        v2f a; a.x = sel; a.y = sel;

        // int-element offset of first int64 col this lane-half needs (k = 2*lh)
        const unsigned cb = (unsigned)(((row0 + m) * 32 + 2 * lh) * 2);

        #pragma unroll
        for (int kk = 0; kk < 8; ++kk) {
            // One 16B-aligned b128 load = low+high dwords of col[k], col[k+1]
            const v4i c = *(const v4i*)(col2 + (cb + 8u * (unsigned)kk));
            const unsigned o0 = (unsigned)(c.x * D) + ncl;  // element offsets
            const unsigned o1 = (unsigned)(c.z * D) + ncl;

            v2f b;
            b.x = X[o0];                    // coalesced 64B per half-wave
            b.y = X[o1];

            if (kk & 1)
                acc1 = __builtin_amdgcn_wmma_f32_16x16x4_f32(
                    false, a, false, b, (short)0, acc1, false, false);
            else
                acc0 = __builtin_amdgcn_wmma_f32_16x16x4_f32(
                    false, a, false, b, (short)0, acc0, false, false);
        }
    }

    v8f acc = acc0 + acc1;

    // C/D layout: VGPR r -> M = r + 8*lh, N = lane & 15
    if (nok) {
        const unsigned ob = (unsigned)((row0 + 8 * lh) * D + nf);
        #pragma unroll
        for (int r = 0; r < 8; ++r)
            out[ob + (unsigned)(r * D)] = acc[r];
    }
}

// ---------------------------------------------------------------------------
// Generic fallback: any degree / any N, 64-bit safe, k- and row-masked.
// ---------------------------------------------------------------------------
__global__ __launch_bounds__(256) void sag_wmma_generic_kernel(
    const float* __restrict__ X,
    const long long* __restrict__ col,
    float* __restrict__ out,
    int N, int D, int deg, int FB, int tiles)
{
    const int lane = threadIdx.x & 31;
    const int wave = threadIdx.x >> 5;
    const int tile = blockIdx.x * 8 + wave;
    if (tile >= tiles) return;

    const int rb   = tile / FB;
    const int fb   = tile % FB;
    const int row0 = rb * 16;
    const int n0   = fb * 16;

    const int lh = lane >> 4;
    const int n  = lane & 15;
    const int nf = n0 + n;
    const int nok = (nf < D);
    const int nc  = nok ? nf : 0;

    v8f acc = {};
    const int kchunks = (deg + 3) >> 2;

    for (int m = 0; m < 16; ++m) {
        const int rrow = row0 + m;
        const int rok  = (rrow < N);
        const float sel = (rok && (n == m)) ? 1.0f : 0.0f;
        v2f a; a.x = sel; a.y = sel;

        const long long ebase = (long long)(rok ? rrow : 0) * (long long)deg;

        for (int kk = 0; kk < kchunks; ++kk) {
            const int k0 = kk * 4 + 2 * lh;
            const int k1 = k0 + 1;
            // Clamp k (load a valid address) and zero the contribution.
            const long long e0 = ebase + (k0 < deg ? k0 : 0);
            const long long e1 = ebase + (k1 < deg ? k1 : 0);
            const float m0 = (k0 < deg) ? 1.0f : 0.0f;
            const float m1 = (k1 < deg) ? 1.0f : 0.0f;
            const long long i0 = col[e0];
            const long long i1 = col[e1];

            v2f b;
            b.x = X[i0 * (long long)D + nc] * m0;
            b.y = X[i1 * (long long)D + nc] * m1;

            acc = __builtin_amdgcn_wmma_f32_16x16x4_f32(
                false, a, false, b, (short)0, acc, false, false);
        }
    }

    if (nok) {
        #pragma unroll
        for (int r = 0; r < 8; ++r) {
            const int orow = row0 + r + 8 * lh;
            if (orow < N)
                out[(long long)orow * (long long)D + nf] = acc[r];
        }
    }
}

extern "C" void kernel_launch(void* const* d_in, const int* in_sizes, int n_in,
                              void* d_out, int out_size, void* d_ws, size_t ws_size,
                              hipStream_t stream) {
    const float*     X   = (const float*)d_in[0];
    const long long* rp  = (const long long*)d_in[1];   // uniform-degree CSR
    const long long* col = (const long long*)d_in[2];
    float*           out = (float*)d_out;
    (void)rp; (void)d_ws; (void)ws_size; (void)n_in; (void)out_size;

    const int N   = in_sizes[1] - 1;          // row_pointers has N+1 entries
    const int D   = in_sizes[0] / N;          // 100
    const int deg = in_sizes[2] / N;          // 32 (uniform)

    const int RB     = (N + 15) / 16;
    const int FB     = (D + 15) / 16;
    const int tiles  = RB * FB;
    const int blocks = (tiles + 7) / 8;       // 8 waves (tiles) per 256-thread block

    const bool fast = (deg == 32) && (N % 16 == 0) &&
                      ((long long)N * D   < 0x7FFFFFFFLL) &&
                      ((long long)N * deg < 0x7FFFFFFFLL);

    if (fast) {
        sag_wmma_deg32_kernel<<<blocks, 256, 0, stream>>>(
            X, (const int*)col, out, D, FB, tiles);
    } else {
        sag_wmma_generic_kernel<<<blocks, 256, 0, stream>>>(
            X, col, out, N, D, deg, FB, tiles);
    }
}